// MaskLoss_77850577207839
// MI455X (gfx1250) — compile-verified
//
#include <hip/hip_runtime.h>
#include <hip/hip_bf16.h>

typedef __attribute__((ext_vector_type(2))) float v2f;
typedef __attribute__((ext_vector_type(4))) float v4f;
typedef __attribute__((ext_vector_type(8))) float v8f;

#define B_   2
#define C_   128
#define N_   3
#define Q_   144     // 12*12 = 9 tiles of 16
#define HW_  4096    // 64*64
#define HT_  256
#define EPS_ 1e-7f
#define WEIGHT_ 3.0f

#if defined(__has_builtin) && __has_builtin(__builtin_amdgcn_tanhf)
__device__ __forceinline__ float sigmoid_fast(float v) {
    // v_tanh_f32: one trans op instead of exp+rcp
    return fmaf(0.5f, __builtin_amdgcn_tanhf(0.5f * v), 0.5f);
}
#else
__device__ __forceinline__ float sigmoid_fast(float v) {
    return 1.0f / (1.0f + __expf(-v));
}
#endif

// ---------------------------------------------------------------------------
// Phase 1: maps[bn][q][hw] = sum_c kern[b][c][n][q] * mask[b][c][n][hw]
// One wave -> one 16(M=q) x 16(N=hw) f32 tile, K=128 via 32x V_WMMA_F32_16X16X4_F32.
// ---------------------------------------------------------------------------
__global__ __launch_bounds__(256)
void MaskLoss_gemm_wmma(const float* __restrict__ mask,
                        const float* __restrict__ kern,
                        float* __restrict__ maps) {
    __shared__ float As[16][130];           // padded pitch: conflict-free column reads
    const int bn  = blockIdx.z;
    const int b   = bn / N_;
    const int n   = bn % N_;
    const int qt  = blockIdx.y;
    const int tid = threadIdx.x;

    for (int i = tid; i < 16 * C_; i += 256) {
        const int r = i >> 7, k = i & (C_ - 1);
        As[r][k] = kern[((size_t)(b * C_ + k) * N_ + n) * Q_ + (qt * 16 + r)];
    }
    __syncthreads();

    const int wave = tid >> 5, lane = tid & 31;
    const int half = lane >> 4;
    const int lrow = lane & 15;
    const int koff = half * 2;              // ISA A/B layout: lane halves hold K pairs
    const int hw   = blockIdx.x * 128 + wave * 16 + lrow;

    const float* bcol = mask + ((size_t)(b * C_) * N_ + n) * HW_ + hw;
    const size_t cstride = (size_t)N_ * HW_;

    v8f acc = {};
#if defined(__has_builtin) && __has_builtin(__builtin_amdgcn_wmma_f32_16x16x4_f32)
#pragma unroll 8
    for (int k = 0; k < C_; k += 4) {
        v2f a, bb;
        a.x  = As[lrow][k + koff];
        a.y  = As[lrow][k + koff + 1];
        bb.x = bcol[(size_t)(k + koff) * cstride];
        bb.y = bcol[(size_t)(k + koff + 1) * cstride];
        acc = __builtin_amdgcn_wmma_f32_16x16x4_f32(
            false, a, false, bb, (short)0, acc, false, false);
    }
#else
    for (int k = 0; k < C_; ++k) {
        const float bv = bcol[(size_t)k * cstride];
#pragma unroll
        for (int r = 0; r < 8; ++r)
            acc[r] = fmaf(As[r + 8 * half][k], bv, acc[r]);
    }
#endif

    float* dst = maps + ((size_t)bn * Q_ + qt * 16) * HW_ + hw;
#pragma unroll
    for (int r = 0; r < 8; ++r)
        dst[(size_t)(r + 8 * half) * HW_] = acc[r];
}

// ---------------------------------------------------------------------------
// Phase 2: fused x4 bilinear upsample + sigmoid + dice partial sums.
// One block per (b,n,q). Exploits the fixed x4 ratio: interpolation weights
// are compile-time constants {0.625,0.875,0.125,0.375}; index clamping
// reproduces align_corners=False edges. Thread (tsub=tid>>6, tx=tid&63)
// owns output rows y=4m+tsub and columns x=4tx..4tx+3:
//   - one NT global_load_b128 of target per 4 pixels
//   - 6 LDS reads + shared row-interp per 4 pixels
//   - one v_tanh per pixel for sigmoid
// ---------------------------------------------------------------------------
__global__ __launch_bounds__(256)
void MaskLoss_dice_partial(const float* __restrict__ maps,
                           const float* __restrict__ target,
                           float* __restrict__ partials) {
    __shared__ float tile[64][68];          // 272B pitch: 16B-aligned rows
    __shared__ float red[3][8];
    const int bnq = blockIdx.x;             // bn*Q_ + q
    const int q   = bnq % Q_;
    const int bn  = bnq / Q_;
    const int b   = bn / N_;
    const int n   = bn % N_;
    const int tid = threadIdx.x;

    // Stage 64x64 maps tile into LDS with b128 transfers.
    const v4f* src4 = (const v4f*)(maps + (size_t)bnq * HW_);
    for (int i = tid; i < HW_ / 4; i += 256) {
        const int r = i >> 4, c4 = i & 15;
        *(v4f*)&tile[r][c4 * 4] = src4[i];
    }
    __syncthreads();

    const int tsub = tid >> 6;              // output row within group of 4 (wave-uniform)
    const int tx   = tid & 63;              // owns x = 4tx .. 4tx+3
    const int cm1  = (tx > 0) ? tx - 1 : 0;
    const int kc   = tx;
    const int kp1  = (tx < 63) ? tx + 1 : 63;
    const int roff = (tsub >= 2) ? 0 : -1;
    const float wr = 0.625f + 0.25f * (float)tsub - ((tsub >= 2) ? 1.0f : 0.0f);

    const v4f* tgt4 = (const v4f*)(target + ((((size_t)b * Q_ + q) * N_ + n) << 16));

    float spt = 0.0f, spp = 0.0f, st = 0.0f;
    for (int m = 0; m < 64; ++m) {          // y = 4m + tsub
        const int r0 = (m + roff > 0) ? (m + roff) : 0;
        const int r1 = (m + roff + 1 < 63) ? (m + roff + 1) : 63;

        const float am1 = tile[r0][cm1], a0 = tile[r0][kc], ap1 = tile[r0][kp1];
        const float bm1 = tile[r1][cm1], b0 = tile[r1][kc], bp1 = tile[r1][kp1];
        const float gm1 = fmaf(bm1 - am1, wr, am1);   // row-interp, shared by 4 cols
        const float g0  = fmaf(b0  - a0,  wr, a0);
        const float gp1 = fmaf(bp1 - ap1, wr, ap1);

        const float p0 = sigmoid_fast(fmaf(g0  - gm1, 0.625f, gm1));
        const float p1 = sigmoid_fast(fmaf(g0  - gm1, 0.875f, gm1));
        const float p2 = sigmoid_fast(fmaf(gp1 - g0,  0.125f, g0));
        const float p3 = sigmoid_fast(fmaf(gp1 - g0,  0.375f, g0));

        const v4f t = __builtin_nontemporal_load(&tgt4[(((m << 2) + tsub) << 6) + tx]);

        spt = fmaf(p0, t.x, spt); spt = fmaf(p1, t.y, spt);
        spt = fmaf(p2, t.z, spt); spt = fmaf(p3, t.w, spt);
        spp = fmaf(p0, p0, spp);  spp = fmaf(p1, p1, spp);
        spp = fmaf(p2, p2, spp);  spp = fmaf(p3, p3, spp);
        st += (t.x + t.y) + (t.z + t.w);
    }

    // wave32 reduction, then cross-wave via LDS
#pragma unroll
    for (int off = 16; off > 0; off >>= 1) {
        spt += __shfl_down(spt, off, 32);
        spp += __shfl_down(spp, off, 32);
        st  += __shfl_down(st,  off, 32);
    }
    const int wave = tid >> 5, lane = tid & 31;
    if (lane == 0) { red[0][wave] = spt; red[1][wave] = spp; red[2][wave] = st; }
    __syncthreads();
    if (tid == 0) {
        float a = 0.f, c = 0.f, d = 0.f;
#pragma unroll
        for (int w = 0; w < 8; ++w) { a += red[0][w]; c += red[1][w]; d += red[2][w]; }
        partials[bnq * 3 + 0] = a;   // sum(pred*tgt)
        partials[bnq * 3 + 1] = c;   // sum(pred*pred)
        partials[bnq * 3 + 2] = d;   // sum(tgt) == sum(tgt^2), binary
    }
}

// ---------------------------------------------------------------------------
// Phase 3: 864 -> scalar loss (mirrors the reference reduction exactly).
// ---------------------------------------------------------------------------
__global__ void MaskLoss_finalize(const float* __restrict__ partials,
                                  float* __restrict__ out) {
    if (threadIdx.x != 0 || blockIdx.x != 0) return;
    float n_loss_sum = 0.0f;
    float frames = 0.0f;
    for (int n = 0; n < N_; ++n) {
        float nden = 0.0f, bsum = 0.0f, anyv = 0.0f;
        for (int b = 0; b < B_; ++b) {
            float cnt = 0.0f, acc = 0.0f;
            for (int q = 0; q < Q_; ++q) {
                const int bnq = (b * N_ + n) * Q_ + q;
                const float pt = partials[bnq * 3 + 0];
                const float pp = partials[bnq * 3 + 1];
                const float t  = partials[bnq * 3 + 2];
                const float valid = (t > 0.0f) ? 1.0f : 0.0f;
                const float dice  = 1.0f - 2.0f * pt / (pp + t + EPS_);
                cnt += valid;
                acc += dice * valid;
            }
            bsum += acc / fmaxf(cnt, 1.0f);
            if (cnt > 0.0f) { nden += 1.0f; anyv = 1.0f; }
        }
        n_loss_sum += bsum / fmaxf(nden, 1.0f);
        frames += anyv;
    }
    out[0] = n_loss_sum / fmaxf(frames, 1.0f) * WEIGHT_;
}

// ---------------------------------------------------------------------------
extern "C" void kernel_launch(void* const* d_in, const int* in_sizes, int n_in,
                              void* d_out, int out_size, void* d_ws, size_t ws_size,
                              hipStream_t stream) {
    const float* mask   = (const float*)d_in[0];   // [B,C,N,64,64]
    const float* kern   = (const float*)d_in[1];   // [B,C,N,12,12]
    const float* target = (const float*)d_in[2];   // [B,Q,N,256,256]

    float* maps     = (float*)d_ws;                         // 14.2 MB
    float* partials = maps + (size_t)B_ * N_ * Q_ * HW_;    // 864*3 floats

    dim3 g1(HW_ / 128, Q_ / 16, B_ * N_);                   // (32, 9, 6)
    MaskLoss_gemm_wmma<<<g1, 256, 0, stream>>>(mask, kern, maps);

    MaskLoss_dice_partial<<<B_ * N_ * Q_, 256, 0, stream>>>(maps, target, partials);

    MaskLoss_finalize<<<1, 32, 0, stream>>>(partials, (float*)d_out);
}